// NBVLoss_26869315404229
// MI455X (gfx1250) — compile-verified
//
#include <hip/hip_runtime.h>

// Weighted BCE loss reduction for MI455X (gfx1250, wave32).
// Memory-bound: 2 x 67.1M x 4B = ~537 MB streamed @ 23.3 TB/s => ~23 us floor.
// - Non-temporal global_load_b128 (pure streaming; no L2 reuse of 537 MB vs 192 MB L2)
// - 2x unrolled grid-stride loop, dual accumulators for load latency hiding
// - Exact wave32 reduction via V_WMMA_F32_16X16X4_F32 + one __shfl_down
// - Deterministic two-pass reduction (no float atomics)

typedef __attribute__((ext_vector_type(2))) float v2f;
typedef __attribute__((ext_vector_type(4))) float vf4;   // native vector for NT loads
typedef __attribute__((ext_vector_type(8))) float v8f;

#define LAMBDA_FOR0 0.4f
#define LAMBDA_FOR1 1.6f
#define LOG_CLAMP  -100.0f

#define BLOCK_THREADS 256
#define GRID_BLOCKS   2048
#define WAVES_PER_BLOCK (BLOCK_THREADS / 32)

// Exact wave32 sum via one WMMA:
//   A(16x4): lane L<16 -> row L, {K0=acc, K1=0}; lane L>=16 -> row L-16, {K2=acc, K3=0}
//   B(4x16) = ones  =>  D[m][n] = acc_m + acc_{m+16}  (same for every column n)
//   C/D layout: lane L<16 holds D[v][L] in VGPR v; lane L>=16 holds D[v+8][L-16].
//   sum of lane n's 8 regs + lane (n+16)'s 8 regs = full wave sum.
// Requires EXEC == all ones (callers guarantee convergence).
__device__ __forceinline__ float wave_reduce_wmma(float acc) {
    v2f a;
    a[0] = acc;
    a[1] = 0.0f;
    v2f b;
    b[0] = 1.0f;
    b[1] = 1.0f;
    v8f c = {};
    c = __builtin_amdgcn_wmma_f32_16x16x4_f32(
        /*neg_a=*/false, a, /*neg_b=*/false, b,
        /*c_mod=*/(short)0, c, /*reuse_a=*/false, /*reuse_b=*/false);
    float s = c[0] + c[1] + c[2] + c[3] + c[4] + c[5] + c[6] + c[7];
    s += __shfl_down(s, 16, 32);   // lanes 0..15 now hold the exact wave total
    return s;
}

// target is exactly 0.0f or 1.0f:
//   t==1: -LAMBDA_FOR1 * max(log(p),   -100)
//   t==0: -LAMBDA_FOR0 * max(log(1-p), -100)
// Select the log argument so only one transcendental is issued per element.
__device__ __forceinline__ float bce_term(float p, float t) {
    const bool one = (t != 0.0f);
    const float x  = one ? p : (1.0f - p);
    const float lg = fmaxf(__logf(x), LOG_CLAMP);
    const float w  = one ? LAMBDA_FOR1 : LAMBDA_FOR0;
    return -w * lg;
}

__device__ __forceinline__ float bce_term4(vf4 pv, vf4 tv) {
    return bce_term(pv[0], tv[0]) + bce_term(pv[1], tv[1]) +
           bce_term(pv[2], tv[2]) + bce_term(pv[3], tv[3]);
}

__global__ void __launch_bounds__(BLOCK_THREADS)
nbv_loss_partial(const vf4* __restrict__ pred4,
                 const vf4* __restrict__ targ4,
                 const float* __restrict__ pred,
                 const float* __restrict__ targ,
                 float* __restrict__ block_partial,
                 long long n4, long long n) {
    const long long stride = (long long)gridDim.x * BLOCK_THREADS;
    long long i = (long long)blockIdx.x * BLOCK_THREADS + threadIdx.x;

    float acc0 = 0.0f;
    float acc1 = 0.0f;

    // 2x unrolled: 4 independent NT b128 loads in flight per iteration.
    for (; i + stride < n4; i += 2 * stride) {
        const vf4 p0 = __builtin_nontemporal_load(&pred4[i]);
        const vf4 t0 = __builtin_nontemporal_load(&targ4[i]);
        const vf4 p1 = __builtin_nontemporal_load(&pred4[i + stride]);
        const vf4 t1 = __builtin_nontemporal_load(&targ4[i + stride]);
        acc0 += bce_term4(p0, t0);
        acc1 += bce_term4(p1, t1);
    }
    if (i < n4) {
        const vf4 p0 = __builtin_nontemporal_load(&pred4[i]);
        const vf4 t0 = __builtin_nontemporal_load(&targ4[i]);
        acc0 += bce_term4(p0, t0);
    }

    // Scalar tail (n not divisible by 4) — global thread 0 only; reconverges
    // before the WMMA reduce below.
    if (blockIdx.x == 0 && threadIdx.x == 0) {
        for (long long k = n4 * 4; k < n; ++k) {
            acc0 += bce_term(pred[k], targ[k]);
        }
    }

    // Full-EXEC wave reduction via WMMA, then cross-wave via LDS.
    const float wsum = wave_reduce_wmma(acc0 + acc1);

    __shared__ float smem[WAVES_PER_BLOCK];
    const int lane = threadIdx.x & 31;
    const int wave = threadIdx.x >> 5;
    if (lane == 0) smem[wave] = wsum;
    __syncthreads();

    if (threadIdx.x == 0) {
        float bsum = 0.0f;
        #pragma unroll
        for (int w = 0; w < WAVES_PER_BLOCK; ++w) bsum += smem[w];
        block_partial[blockIdx.x] = bsum;
    }
}

__global__ void __launch_bounds__(BLOCK_THREADS)
nbv_loss_final(const float* __restrict__ block_partial,
               int n_partials,
               float* __restrict__ out) {
    float acc = 0.0f;
    for (int i = threadIdx.x; i < n_partials; i += BLOCK_THREADS) {
        acc += block_partial[i];
    }

    const float wsum = wave_reduce_wmma(acc);

    __shared__ float smem[WAVES_PER_BLOCK];
    const int lane = threadIdx.x & 31;
    const int wave = threadIdx.x >> 5;
    if (lane == 0) smem[wave] = wsum;
    __syncthreads();

    if (threadIdx.x == 0) {
        float total = 0.0f;
        #pragma unroll
        for (int w = 0; w < WAVES_PER_BLOCK; ++w) total += smem[w];
        out[0] = total;
    }
}

extern "C" void kernel_launch(void* const* d_in, const int* in_sizes, int n_in,
                              void* d_out, int out_size, void* d_ws, size_t ws_size,
                              hipStream_t stream) {
    const float* pred = (const float*)d_in[0];
    const float* targ = (const float*)d_in[1];
    float* out = (float*)d_out;
    float* block_partial = (float*)d_ws;   // GRID_BLOCKS floats of scratch

    const long long n  = (long long)in_sizes[0];
    const long long n4 = n >> 2;

    nbv_loss_partial<<<GRID_BLOCKS, BLOCK_THREADS, 0, stream>>>(
        (const vf4*)pred, (const vf4*)targ, pred, targ,
        block_partial, n4, n);

    nbv_loss_final<<<1, BLOCK_THREADS, 0, stream>>>(
        block_partial, GRID_BLOCKS, out);
}